// Sampler_64218351010122
// MI455X (gfx1250) — compile-verified
//
#include <hip/hip_runtime.h>
#include <cstdint>

// ---------------------------------------------------------------------------
// Sampler for B=128, V=128256.
// Phase 1: one block per (row, chunk): async global->LDS staging of logits,
//          fused penalties + greedy argmax + exact top-32 of the 32064-elem
//          chunk. Selection state is packed (value,index) u64 keys; wave 0
//          performs the 32 extraction rounds barrier-free with cooperative
//          segment rescans.
// Phase 2: one block per row (128 threads): temperature scale, softmax,
//          top-k / top-p filtering with exact reference tie semantics,
//          final softmax, argmax(prob/q) sampling, greedy override.
// Workspace usage: 135,168 bytes of d_ws.  LDS: 143,360 B / block.
// ---------------------------------------------------------------------------

#define B_ROWS     128
#define VOCAB      128256
#define NCH        4
#define CSZ        32064            // VOCAB / NCH
#define KTOP       32
#define CAND       128              // NCH * KTOP
#define P1_THREADS 512
#define SEG        64               // logical chunk elements per phase-1 thread
#define VPAD       68               // padded LDS stride (dwords): kills 64-bank conflicts
#define NEG_INF    (-__builtin_inff())

// ---- monotonic (value,index) packing: bigger key == (bigger val, smaller idx)
__device__ __forceinline__ uint32_t ford(float f) {
  uint32_t u = __float_as_uint(f);
  uint32_t m = (uint32_t)((int32_t)u >> 31) | 0x80000000u;
  return u ^ m;
}
__device__ __forceinline__ float finv(uint32_t o) {
  uint32_t m = (o & 0x80000000u) ? 0x80000000u : 0xFFFFFFFFu;
  return __uint_as_float(o ^ m);
}
__device__ __forceinline__ unsigned long long packkey(float f, int pos) {
  return ((unsigned long long)ford(f) << 32) | (unsigned long long)(0xFFFFFFFFu - (uint32_t)pos);
}
__device__ __forceinline__ int keypos(unsigned long long k) {
  return (int)(0xFFFFFFFFu - (uint32_t)k);
}
__device__ __forceinline__ unsigned long long kmax(unsigned long long a, unsigned long long b) {
  return a > b ? a : b;
}

// ---- gfx1250 async global->LDS copy (ASYNCcnt path); INST_OFFSET applies to
//      both the LDS and global side, so one base address pair serves 16 copies.
template <int OFF>
__device__ __forceinline__ void async_copy_b128(const void* gsrc, uint32_t lds_byte_addr) {
  asm volatile("global_load_async_to_lds_b128 %0, %1, off offset:%2"
               :: "v"(lds_byte_addr), "v"(gsrc), "n"(OFF) : "memory");
}
__device__ __forceinline__ void wait_async0() {
#if __has_builtin(__builtin_amdgcn_s_wait_asynccnt)
  __builtin_amdgcn_s_wait_asynccnt(0);
#else
  asm volatile("s_wait_asynccnt 0" ::: "memory");
#endif
}

// ---------------------------------------------------------------------------
// Phase 1
// ---------------------------------------------------------------------------
__global__ __launch_bounds__(P1_THREADS)
void sampler_phase1(const float* __restrict__ logits,
                    const float* __restrict__ counts,
                    const unsigned char* __restrict__ pmask,
                    const float* __restrict__ presence,
                    const float* __restrict__ frequency,
                    const float* __restrict__ repetition,
                    float* __restrict__ topv, int* __restrict__ topi,
                    float* __restrict__ gval, int* __restrict__ gidx)
{
  extern __shared__ unsigned char smem[];
  float* vals = (float*)smem;                                   // P1_THREADS*VPAD dwords
  unsigned long long* lkey =
      (unsigned long long*)(smem + (size_t)P1_THREADS * VPAD * 4); // P1_THREADS keys

  const int row = (int)(blockIdx.x >> 2);
  const int ch  = (int)(blockIdx.x & 3);
  const int t   = (int)threadIdx.x;
  const size_t base = (size_t)row * VOCAB + (size_t)ch * CSZ;

  const bool active = (t * SEG) < CSZ;   // threads 0..500 own 64 elements each

  // ---- stage raw logits chunk into LDS via async copies -------------------
  if (active) {
    const float* gsrc = logits + base + (size_t)t * SEG;
    uint32_t l0 = (uint32_t)(uintptr_t)(vals + (size_t)t * VPAD);
    __builtin_prefetch(counts + base + (size_t)t * SEG, 0, 3);
    __builtin_prefetch(pmask  + base + (size_t)t * SEG, 0, 3);
    async_copy_b128<0>(gsrc, l0);    async_copy_b128<16>(gsrc, l0);
    async_copy_b128<32>(gsrc, l0);   async_copy_b128<48>(gsrc, l0);
    async_copy_b128<64>(gsrc, l0);   async_copy_b128<80>(gsrc, l0);
    async_copy_b128<96>(gsrc, l0);   async_copy_b128<112>(gsrc, l0);
    async_copy_b128<128>(gsrc, l0);  async_copy_b128<144>(gsrc, l0);
    async_copy_b128<160>(gsrc, l0);  async_copy_b128<176>(gsrc, l0);
    async_copy_b128<192>(gsrc, l0);  async_copy_b128<208>(gsrc, l0);
    async_copy_b128<224>(gsrc, l0);  async_copy_b128<240>(gsrc, l0);
  }
  wait_async0();
  __syncthreads();

  const float rep    = repetition[row];
  const float repinv = 1.0f / rep;
  const float fr     = frequency[row];
  const float pr     = presence[row];

  unsigned long long gkey = 0ull;   // greedy (row-global) partial
  unsigned long long skey = 0ull;   // per-thread segment max

  if (active) {
    const float4*   cp = (const float4*)  (counts + base + (size_t)t * SEG);
    const uint32_t* mp = (const uint32_t*)(pmask  + base + (size_t)t * SEG);
    float4* vb = (float4*)(vals + (size_t)t * VPAD);
#pragma unroll 4
    for (int g = 0; g < SEG / 4; ++g) {
      float4   l4 = vb[g];
      float4   c4 = cp[g];
      uint32_t mw = mp[g];
      float ls[4] = {l4.x, l4.y, l4.z, l4.w};
      float cs[4] = {c4.x, c4.y, c4.z, c4.w};
      float rs[4];
#pragma unroll
      for (int e = 0; e < 4; ++e) {
        int   pos = t * SEG + g * 4 + e;       // chunk-local logical index
        float l   = ls[e];
        float c   = cs[e];
        float occ = fminf(c, 1.0f);            // counts - max(counts-1,0)
        float pf  = ((mw >> (8 * e)) & 0xffu) ? 1.0f : 0.0f;
        float pen = (l > 0.0f) ? l * repinv : l * rep;
        float l2  = ((occ + pf) > 0.0f) ? pen : l;
        float l3  = l2 - fr * c - pr * occ;
        rs[e] = l3;
        skey = kmax(skey, packkey(l3, pos));
        gkey = kmax(gkey, packkey(l3, ch * CSZ + pos));
      }
      vb[g] = make_float4(rs[0], rs[1], rs[2], rs[3]);
    }
  }
  lkey[t] = skey;
  __syncthreads();

  // ---- wave 0 extracts top-32 (descending, stable by index), no barriers --
  if (t < 32) {
    for (int r = 0; r < KTOP; ++r) {
      // scan the 512 cached segment keys, conflict-free (lane t -> t+32k)
      unsigned long long bk = 0ull;
#pragma unroll
      for (int k = 0; k < P1_THREADS / 32; ++k)
        bk = kmax(bk, lkey[t + 32 * k]);
#pragma unroll
      for (int off = 16; off > 0; off >>= 1)
        bk = kmax(bk, (unsigned long long)__shfl_xor((unsigned long long)bk, off));

      int pos = keypos(bk);                    // chunk-local winner
      if (t == 0) {
        topv[row * CAND + ch * KTOP + r] = finv((uint32_t)(bk >> 32));
        topi[row * CAND + ch * KTOP + r] = ch * CSZ + pos;
      }
      int o  = pos >> 6;                       // owning segment
      int eo = pos & (SEG - 1);
      if (t == (eo >> 1)) vals[o * VPAD + eo] = NEG_INF;   // zap (in-order DS)

      // cooperative rescan of segment o: 2 elements per lane
      float2 xy = *(const float2*)(vals + o * VPAD + 2 * t);
      unsigned long long nk = kmax(packkey(xy.x, o * SEG + 2 * t),
                                   packkey(xy.y, o * SEG + 2 * t + 1));
#pragma unroll
      for (int off = 16; off > 0; off >>= 1)
        nk = kmax(nk, (unsigned long long)__shfl_xor((unsigned long long)nk, off));
      if (t == 0) lkey[o] = nk;
    }
  }
  __syncthreads();          // extraction done before lkey reuse

  // ---- reduce greedy argmax partial for this chunk ------------------------
  lkey[t] = gkey;
  __syncthreads();
  if (t < 32) {
    unsigned long long bk = 0ull;
#pragma unroll
    for (int k = 0; k < P1_THREADS / 32; ++k)
      bk = kmax(bk, lkey[t + 32 * k]);
#pragma unroll
    for (int off = 16; off > 0; off >>= 1)
      bk = kmax(bk, (unsigned long long)__shfl_xor((unsigned long long)bk, off));
    if (t == 0) {
      gval[row * NCH + ch] = finv((uint32_t)(bk >> 32));
      gidx[row * NCH + ch] = keypos(bk);
    }
  }
}

// ---------------------------------------------------------------------------
// Phase 2
// ---------------------------------------------------------------------------
__global__ __launch_bounds__(CAND)
void sampler_phase2(const float* __restrict__ topv, const int* __restrict__ topi,
                    const float* __restrict__ gval, const int* __restrict__ gidx,
                    const float* __restrict__ temperature,
                    const int* __restrict__ top_k,
                    const float* __restrict__ top_p,
                    const float* __restrict__ q_samples,
                    float* __restrict__ out)
{
  __shared__ float sp[CAND];
  __shared__ float ssort[CAND];
  __shared__ float rf[CAND];
  __shared__ int   ri[CAND];

  const int b = (int)blockIdx.x;
  const int i = (int)threadIdx.x;

  const float temp = temperature[b];
  const float te   = (temp < 1e-5f) ? 1.0f : temp;
  const float v    = topv[b * CAND + i] / te;
  const int   idx  = topi[b * CAND + i];

  // softmax over 128 candidate logits
  rf[i] = v; __syncthreads();
  for (int s = CAND / 2; s > 0; s >>= 1) { if (i < s) rf[i] = fmaxf(rf[i], rf[i + s]); __syncthreads(); }
  float m = rf[0]; __syncthreads();
  float e1 = expf(v - m);
  rf[i] = e1; __syncthreads();
  for (int s = CAND / 2; s > 0; s >>= 1) { if (i < s) rf[i] += rf[i + s]; __syncthreads(); }
  float ssum = rf[0]; __syncthreads();
  float prob = e1 / ssum;
  sp[i] = prob; __syncthreads();

  // stable ascending rank (ties broken by original index) -> sorted probs
  int rank = 0;
  for (int j = 0; j < CAND; ++j) {
    float pj = sp[j];
    rank += (pj < prob || (pj == prob && j < i)) ? 1 : 0;
  }
  ssort[rank] = prob;
  __syncthreads();

  // left-to-right cumsum at sorted position i (matches jnp.cumsum)
  float c = 0.0f;
  for (int j = 0; j <= i; ++j) c += ssort[j];
  const float p = top_p[b];
  int maskbit = ((c <= 1.0f - p) && (i != CAND - 1)) ? 1 : 0;
  ri[i] = maskbit; __syncthreads();
  for (int s = CAND / 2; s > 0; s >>= 1) { if (i < s) ri[i] += ri[i + s]; __syncthreads(); }
  int count_p = ri[0]; __syncthreads();
  float cutoff_p = ssort[count_p];

  const int k = top_k[b];
  float cutoff_k;
  if (k <= 0 || k >= VOCAB) {
    cutoff_k = NEG_INF;
  } else {
    int ck = CAND - k;
    ck = ck < 0 ? 0 : (ck > CAND - 1 ? CAND - 1 : ck);
    cutoff_k = ssort[ck];
  }
  float cut  = fmaxf(cutoff_p, cutoff_k);
  float filt = (prob < cut) ? NEG_INF : v;

  // softmax over filtered values
  rf[i] = filt; __syncthreads();
  for (int s = CAND / 2; s > 0; s >>= 1) { if (i < s) rf[i] = fmaxf(rf[i], rf[i + s]); __syncthreads(); }
  float m2 = rf[0]; __syncthreads();
  float e2 = (filt == NEG_INF) ? 0.0f : expf(filt - m2);
  rf[i] = e2; __syncthreads();
  for (int s = CAND / 2; s > 0; s >>= 1) { if (i < s) rf[i] += rf[i + s]; __syncthreads(); }
  float s2 = rf[0]; __syncthreads();
  float pout = e2 / s2;
  out[CAND + b * CAND + i] = pout;           // probs output

  // argmax(pout / q), first index wins ties
  float q = q_samples[(size_t)b * VOCAB + idx];
  float score = pout / q;
  rf[i] = score; ri[i] = i; __syncthreads();
  for (int s = CAND / 2; s > 0; s >>= 1) {
    if (i < s) {
      float ov = rf[i + s]; int oi = ri[i + s];
      if (ov > rf[i] || (ov == rf[i] && oi < ri[i])) { rf[i] = ov; ri[i] = oi; }
    }
    __syncthreads();
  }
  if (i == 0) {
    int local     = ri[0];
    int sampled_r = topi[b * CAND + local];
    // combine the 4 per-chunk greedy partials (smaller index wins ties)
    float bgv = NEG_INF; int bgi = 0x7fffffff;
    for (int cnum = 0; cnum < NCH; ++cnum) {
      float gv2 = gval[b * NCH + cnum]; int gi2 = gidx[b * NCH + cnum];
      if (gv2 > bgv || (gv2 == bgv && gi2 < bgi)) { bgv = gv2; bgi = gi2; }
    }
    int sampled = (temp < 1e-5f) ? bgi : sampled_r;
    out[b] = (float)sampled;                 // sampled-token output
  }
}

// ---------------------------------------------------------------------------
extern "C" void kernel_launch(void* const* d_in, const int* in_sizes, int n_in,
                              void* d_out, int out_size, void* d_ws, size_t ws_size,
                              hipStream_t stream) {
  const float*         logits      = (const float*)d_in[0];
  const float*         temperature = (const float*)d_in[1];
  const int*           top_k       = (const int*)d_in[2];
  const float*         top_p       = (const float*)d_in[3];
  const float*         counts      = (const float*)d_in[4];
  const unsigned char* pmask       = (const unsigned char*)d_in[5]; // bool bytes
  const float*         presence    = (const float*)d_in[6];
  const float*         frequency   = (const float*)d_in[7];
  const float*         repetition  = (const float*)d_in[8];
  const float*         q_samples   = (const float*)d_in[9];
  float* out = (float*)d_out;

  // workspace layout
  float* topv = (float*)d_ws;                                        // 16384 f
  int*   topi = (int*)((char*)d_ws + (size_t)B_ROWS * CAND * 4);     // 16384 i
  float* gval = (float*)((char*)d_ws + (size_t)B_ROWS * CAND * 8);   // 512 f
  int*   gidx = (int*)((char*)d_ws + (size_t)B_ROWS * CAND * 8 + (size_t)B_ROWS * NCH * 4);

  size_t smem1 = (size_t)P1_THREADS * VPAD * sizeof(float)          // padded chunk
               + (size_t)P1_THREADS * sizeof(unsigned long long);   // segment keys

  sampler_phase1<<<dim3(B_ROWS * NCH), dim3(P1_THREADS), smem1, stream>>>(
      logits, counts, pmask, presence, frequency, repetition,
      topv, topi, gval, gidx);

  sampler_phase2<<<dim3(B_ROWS), dim3(CAND), 0, stream>>>(
      topv, topi, gval, gidx, temperature, top_k, top_p, q_samples, out);
}